// GlobalMeanPool_22849226015146
// MI455X (gfx1250) — compile-verified
//
#include <hip/hip_runtime.h>

typedef float v2f __attribute__((ext_vector_type(2)));
typedef float v8f __attribute__((ext_vector_type(8)));

namespace {
constexpr int kD   = 256;   // feature dim (reference fixed)
constexpr int kSeg = 512;   // NUM_SEGMENTS (reference fixed)
}

__device__ __forceinline__ int lower_bound_i32(const int* __restrict__ b, int n, int key) {
  int lo = 0, hi = n;
  while (lo < hi) {
    int mid = (lo + hi) >> 1;
    if (b[mid] < key) lo = mid + 1; else hi = mid;
  }
  return lo;
}

// Load one 16-row group (4 WMMA B-operands) for this lane's column.
// Lane layout: half = lane>>4 selects rows {+0,+1} vs {+2,+3} of each 4-row
// group (folded into the base pointer); ncol = lane&15 selects the column.
#define LOAD_SET(B0, B1, B2, B3, Q)                                   \
  do {                                                                \
    B0[0] = __builtin_nontemporal_load((Q) + (size_t)0  * kD);        \
    B0[1] = __builtin_nontemporal_load((Q) + (size_t)1  * kD);        \
    B1[0] = __builtin_nontemporal_load((Q) + (size_t)4  * kD);        \
    B1[1] = __builtin_nontemporal_load((Q) + (size_t)5  * kD);        \
    B2[0] = __builtin_nontemporal_load((Q) + (size_t)8  * kD);        \
    B2[1] = __builtin_nontemporal_load((Q) + (size_t)9  * kD);        \
    B3[0] = __builtin_nontemporal_load((Q) + (size_t)12 * kD);        \
    B3[1] = __builtin_nontemporal_load((Q) + (size_t)13 * kD);        \
  } while (0)

// D = ones(16x4) x B(4x16) + D : every D row accumulates the 4-row column sum.
#define WMMA_SET(B0, B1, B2, B3)                                                                        \
  do {                                                                                                  \
    acc0 = __builtin_amdgcn_wmma_f32_16x16x4_f32(false, ones, false, B0, (short)0, acc0, false, false); \
    acc1 = __builtin_amdgcn_wmma_f32_16x16x4_f32(false, ones, false, B1, (short)0, acc1, false, false); \
    acc2 = __builtin_amdgcn_wmma_f32_16x16x4_f32(false, ones, false, B2, (short)0, acc2, false, false); \
    acc3 = __builtin_amdgcn_wmma_f32_16x16x4_f32(false, ones, false, B3, (short)0, acc3, false, false); \
  } while (0)

__global__ __launch_bounds__(256) void seg_mean_wmma_kernel(
    const float* __restrict__ x, const int* __restrict__ batch,
    float* __restrict__ out, int n) {
  const int s     = blockIdx.x;
  const int start = lower_bound_i32(batch, n, s);
  const int end   = lower_bound_i32(batch, n, s + 1);
  const int count = end - start;

  const int tid   = threadIdx.x;
  const int lane  = tid & 31;
  const int wave  = tid >> 5;
  const int half  = lane >> 4;   // rows {r,r+1} vs {r+2,r+3} of each group
  const int ncol  = lane & 15;   // column within this wave's 16-col chunk
  const int cbase = (blockIdx.y * 8 + wave) * 16;

  const float* __restrict__ colp = x + (size_t)start * kD + cbase + ncol;
  const float* __restrict__ p    = colp + (size_t)(2 * half) * kD;

  v2f ones; ones[0] = 1.0f; ones[1] = 1.0f;   // A = 16x4 all-ones
  v8f acc0 = {}; v8f acc1 = {}; v8f acc2 = {}; v8f acc3 = {};

  // -------- software-pipelined main loop: 16 rows / stage, 2 stages deep ----
  const int G = count >> 4;       // number of 16-row groups (uniform per block)
  const float* q = p;
  v2f a0 = {}, a1 = {}, a2 = {}, a3 = {};
  v2f b0 = {}, b1 = {}, b2 = {}, b3 = {};

  if (G > 0) { LOAD_SET(a0, a1, a2, a3, q); q += (size_t)16 * kD; }
  int g = 1;
  for (; g + 1 < G; g += 2) {     // loads for stage g in flight while g-1 computes
    LOAD_SET(b0, b1, b2, b3, q); q += (size_t)16 * kD;
    WMMA_SET(a0, a1, a2, a3);
    LOAD_SET(a0, a1, a2, a3, q); q += (size_t)16 * kD;
    WMMA_SET(b0, b1, b2, b3);
  }
  if (g < G) {                    // one loaded set (a) + one last group to load
    LOAD_SET(b0, b1, b2, b3, q);
    WMMA_SET(a0, a1, a2, a3);
    WMMA_SET(b0, b1, b2, b3);
  } else if (G > 0) {             // only set a left
    WMMA_SET(a0, a1, a2, a3);
  }

  // -------- tail rows (< 16): VALU adds, both lane halves load same column ---
  int r = G << 4;
  float tail = 0.0f;
  for (; r < count; ++r)
    tail += __builtin_nontemporal_load(colp + (size_t)r * kD);

  // Every D row holds the identical column sum; element 0 is valid for both
  // lane halves. IEEE divide to match the reference's sums/denom.
  float total = acc0[0] + acc1[0] + acc2[0] + acc3[0] + tail;
  float denom = (float)(count > 0 ? count : 1);

  if (lane < 16)
    out[(size_t)s * kD + cbase + ncol] = total / denom;
}

extern "C" void kernel_launch(void* const* d_in, const int* in_sizes, int n_in,
                              void* d_out, int out_size, void* d_ws, size_t ws_size,
                              hipStream_t stream) {
  const float* x     = (const float*)d_in[0];
  const int*   batch = (const int*)d_in[1];
  float*       out   = (float*)d_out;
  const int    n     = in_sizes[1];          // number of rows / index entries

  dim3 grid(kSeg, 2);                        // 512 segments x 2 column halves
  seg_mean_wmma_kernel<<<grid, 256, 0, stream>>>(x, batch, out, n);
}